// SageConv_Model_61263413510801
// MI455X (gfx1250) — compile-verified
//
#include <hip/hip_runtime.h>
#include <hip/hip_bf16.h>

typedef float v2f __attribute__((ext_vector_type(2)));
typedef float v8f __attribute__((ext_vector_type(8)));

#define NNODES 50000
#define FEAT 128

// ---------------------------------------------------------------------------
// Zero a float buffer
// ---------------------------------------------------------------------------
__global__ void zero_kernel(float* __restrict__ p, int n) {
  int i = blockIdx.x * blockDim.x + threadIdx.x;
  if (i < n) p[i] = 0.0f;
}

// ---------------------------------------------------------------------------
// Edge aggregation: one wave32 per edge, lane t moves feat[src][4t..4t+3]
// into agg[dst] via f32 atomics. Optionally counts degree (lane 0).
// x/h/agg are L2-resident (25.6 MB each), so this is L2-atomic bound.
// ---------------------------------------------------------------------------
__global__ void edge_agg_kernel(const float* __restrict__ feat,
                                const int* __restrict__ src,
                                const int* __restrict__ dst,
                                int E,
                                float* __restrict__ agg,
                                float* __restrict__ cnt) {
  int gid = blockIdx.x * blockDim.x + threadIdx.x;
  int e = gid >> 5;
  int lane = gid & 31;
  if (e >= E) return;
  int s = src[e];
  int d = dst[e];
  const float4 v =
      *reinterpret_cast<const float4*>(feat + (size_t)s * FEAT + lane * 4);
  float* a = agg + (size_t)d * FEAT + lane * 4;
  atomicAdd(a + 0, v.x);
  atomicAdd(a + 1, v.y);
  atomicAdd(a + 2, v.z);
  atomicAdd(a + 3, v.w);
  if (cnt != nullptr && lane == 0) atomicAdd(cnt + d, 1.0f);
}

// ---------------------------------------------------------------------------
// SAGE layer: out[16 x FOUT] per block-row-tile, one wave per 16-col tile.
//   out = (agg/max(cnt,1)) @ Wl + xin @ Wr + bias   (optional ReLU)
// Uses V_WMMA_F32_16X16X4_F32 (fp32 matrix pipe), K=128 -> 32 k-steps,
// 2 WMMAs per step (mean-path + root-path). EXEC is all-1s throughout.
// A layout (16x4 f32): lanes 0-15 = rows M, v0/v1 = K+0/K+1; lanes 16-31 same
// rows with K+2/K+3. B layout mirrored per column N. C/D: vgpr v = row M=v
// (+8 for upper half), lane&15 = column N.
// ---------------------------------------------------------------------------
template <int FOUT, bool RELU>
__global__ void sage_wmma_kernel(const float* __restrict__ agg,
                                 const float* __restrict__ cnt,
                                 const float* __restrict__ xin,
                                 const float* __restrict__ Wl,
                                 const float* __restrict__ Wr,
                                 const float* __restrict__ bias,
                                 float* __restrict__ out) {
  const int lane = threadIdx.x & 31;
  const int wave = threadIdx.x >> 5;
  const int row0 = blockIdx.x * 16;
  const int col0 = wave * 16;
  const int lrow = lane & 15;          // A row / B,C column within tile
  const int khalf = (lane >> 4) * 2;   // K sub-offset for this lane half
  const int arow = row0 + lrow;

  const float c = cnt[arow];
  const float rinv = 1.0f / fmaxf(c, 1.0f);

  v8f acc = {0.f, 0.f, 0.f, 0.f, 0.f, 0.f, 0.f, 0.f};

#pragma unroll 4
  for (int k = 0; k < FEAT; k += 4) {
    const int kk = k + khalf;
    // mean @ Wl : scale the A fragment by 1/deg on the fly
    float2 am = *reinterpret_cast<const float2*>(agg + (size_t)arow * FEAT + kk);
    v2f a;
    a.x = am.x * rinv;
    a.y = am.y * rinv;
    v2f b;
    b.x = Wl[(size_t)kk * FOUT + col0 + lrow];
    b.y = Wl[(size_t)(kk + 1) * FOUT + col0 + lrow];
    acc = __builtin_amdgcn_wmma_f32_16x16x4_f32(false, a, false, b, (short)0,
                                                acc, false, false);
    // xin @ Wr (root path)
    float2 ax = *reinterpret_cast<const float2*>(xin + (size_t)arow * FEAT + kk);
    v2f a2;
    a2.x = ax.x;
    a2.y = ax.y;
    v2f b2;
    b2.x = Wr[(size_t)kk * FOUT + col0 + lrow];
    b2.y = Wr[(size_t)(kk + 1) * FOUT + col0 + lrow];
    acc = __builtin_amdgcn_wmma_f32_16x16x4_f32(false, a2, false, b2, (short)0,
                                                acc, false, false);
  }

  const float bv = bias[col0 + lrow];
  const int rbase = row0 + ((lane >> 4) * 8);
#pragma unroll
  for (int v = 0; v < 8; ++v) {
    float val = acc[v] + bv;
    if (RELU) val = fmaxf(val, 0.0f);
    out[(size_t)(rbase + v) * FOUT + col0 + lrow] = val;
  }
}

// ---------------------------------------------------------------------------
// Link prediction: logits = [z[ps] ; z[pd]] @ Wlin + blin -> sigmoid
// z (12.8 MB) is L2 resident; one thread per pair, float4 dot products.
// ---------------------------------------------------------------------------
__global__ void link_pred_kernel(const float* __restrict__ z,
                                 const int* __restrict__ ps,
                                 const int* __restrict__ pd,
                                 const float* __restrict__ Wlin,
                                 const float* __restrict__ blin,
                                 float* __restrict__ out, int P) {
  int p = blockIdx.x * blockDim.x + threadIdx.x;
  if (p >= P) return;
  const float* za = z + (size_t)ps[p] * 64;
  const float* zb = z + (size_t)pd[p] * 64;
  float acc = 0.0f;
#pragma unroll
  for (int j = 0; j < 64; j += 4) {
    float4 a = *reinterpret_cast<const float4*>(za + j);
    float4 wa = *reinterpret_cast<const float4*>(Wlin + j);
    acc += a.x * wa.x + a.y * wa.y + a.z * wa.z + a.w * wa.w;
    float4 b = *reinterpret_cast<const float4*>(zb + j);
    float4 wb = *reinterpret_cast<const float4*>(Wlin + 64 + j);
    acc += b.x * wb.x + b.y * wb.y + b.z * wb.z + b.w * wb.w;
  }
  acc += blin[0];
  out[p] = 1.0f / (1.0f + expf(-acc));
}

// ---------------------------------------------------------------------------
extern "C" void kernel_launch(void* const* d_in, const int* in_sizes, int n_in,
                              void* d_out, int out_size, void* d_ws,
                              size_t ws_size, hipStream_t stream) {
  const float* x    = (const float*)d_in[0];
  const int*   ei   = (const int*)d_in[1];
  const int*   ep   = (const int*)d_in[2];
  const float* W1l  = (const float*)d_in[3];
  const float* b1   = (const float*)d_in[4];
  const float* W1r  = (const float*)d_in[5];
  const float* W2l  = (const float*)d_in[6];
  const float* b2   = (const float*)d_in[7];
  const float* W2r  = (const float*)d_in[8];
  const float* Wlin = (const float*)d_in[9];
  const float* blin = (const float*)d_in[10];
  float* out = (float*)d_out;

  const int N = NNODES;
  const int E = in_sizes[1] / 2;
  const int P = in_sizes[2] / 2;
  const int* src = ei;
  const int* dst = ei + E;
  const int* ps  = ep;
  const int* pd  = ep + P;

  // Workspace layout: cnt[N] | agg[N*128] | h[N*128] | z[N*64]  (~64 MB)
  float* cnt = (float*)d_ws;
  float* agg = cnt + N;
  float* h   = agg + (size_t)N * FEAT;
  float* z   = h + (size_t)N * FEAT;

  // 1. zero cnt + agg (contiguous)
  {
    int n = N * (FEAT + 1);
    zero_kernel<<<(n + 255) / 256, 256, 0, stream>>>(cnt, n);
  }
  // 2. aggregate x into agg, count degrees
  {
    long long thr = (long long)E * 32;
    edge_agg_kernel<<<(int)((thr + 255) / 256), 256, 0, stream>>>(
        x, src, dst, E, agg, cnt);
  }
  // 3. layer 1: h = relu(mean @ W1l + x @ W1r + b1)   [N,128]
  sage_wmma_kernel<128, true><<<N / 16, 256, 0, stream>>>(agg, cnt, x, W1l,
                                                          W1r, b1, h);
  // 4. re-zero agg for layer 2
  {
    int n = N * FEAT;
    zero_kernel<<<(n + 255) / 256, 256, 0, stream>>>(agg, n);
  }
  // 5. aggregate h into agg (same degrees)
  {
    long long thr = (long long)E * 32;
    edge_agg_kernel<<<(int)((thr + 255) / 256), 256, 0, stream>>>(
        h, src, dst, E, agg, nullptr);
  }
  // 6. layer 2: z = mean @ W2l + h @ W2r + b2   [N,64]
  sage_wmma_kernel<64, false><<<N / 16, 128, 0, stream>>>(agg, cnt, h, W2l,
                                                          W2r, b2, z);
  // 7. link prediction head
  link_pred_kernel<<<(P + 255) / 256, 256, 0, stream>>>(z, ps, pd, Wlin, blin,
                                                        out, P);
}